// Balanced_sinkhorn_75411035783866
// MI455X (gfx1250) — compile-verified
//
#include <hip/hip_runtime.h>
#include <hip/hip_bf16.h>
#include <math.h>
#include <stdint.h>
#include <stddef.h>

// ---------------------------------------------------------------------------
// Balanced Sinkhorn (B=16384, K=256), f64 internals, E cached once (L2-resident).
// - Fused row+column passes over E, staged via CDNA5 async global->LDS copies
//   (global_load_async_to_lds_b64/b32 + s_wait_asynccnt double buffering).
// - Last-block in-kernel tail reductions remove all intermediate reduce kernels.
// - WMMA inapplicable: all heavy math is f64 GEMV (E spans ~e^±100).
// ---------------------------------------------------------------------------

namespace {

constexpr int    BN  = 16384;          // batch rows
constexpr int    KN  = 256;            // clusters (== blockDim)
constexpr int    NWG = 256;            // blocks in big passes
constexpr int    RPW = BN / NWG;       // 64 rows per block
constexpr int    CHR = 8;              // rows per async chunk (== waves/block)
constexpr int    NCH = RPW / CHR;      // 8 chunks
constexpr double BND = 16384.0;

// ---- CDNA5 async global->LDS copies + split wait counters ------------------
__device__ __forceinline__ void async_b64(unsigned l, const double* g) {
  asm volatile("global_load_async_to_lds_b64 %0, %1, off" :: "v"(l), "v"(g) : "memory");
}
__device__ __forceinline__ void async_b32(unsigned l, const float* g) {
  asm volatile("global_load_async_to_lds_b32 %0, %1, off" :: "v"(l), "v"(g) : "memory");
}
__device__ __forceinline__ void wait_async_0()  { asm volatile("s_wait_asynccnt 0"  ::: "memory"); }
__device__ __forceinline__ void wait_async_8()  { asm volatile("s_wait_asynccnt 8"  ::: "memory"); }
__device__ __forceinline__ void wait_async_16() { asm volatile("s_wait_asynccnt 16" ::: "memory"); }

__device__ __forceinline__ unsigned lds_addr_of(const void* p) {
  return (unsigned)(uintptr_t)p;       // low 32 bits of a generic LDS pointer
}

// ---- last-block election: true for exactly the final arriving block --------
__device__ __forceinline__ bool block_is_last(unsigned* ctr) {
  __shared__ unsigned lastf;
  __threadfence();                     // publish this block's partial stores
  __syncthreads();
  if (threadIdx.x == 0) {
    const unsigned prev = atomicAdd(ctr, 1u);
    lastf = (prev == (unsigned)(NWG - 1)) ? 1u : 0u;
  }
  __syncthreads();
  if (lastf) __threadfence();          // acquire: see all blocks' partials
  return lastf != 0u;
}

// ---- fixed-order stage-2 reductions (run by the last block only) -----------
__device__ __forceinline__ void tail_reduce_fwd(const double* __restrict__ part,
                                                const double* __restrict__ K2,
                                                double* __restrict__ s_out,
                                                double* __restrict__ r_out) {
  const int k = threadIdx.x;
  double a = 0.0;
  for (int w = 0; w < NWG; ++w) a += part[(size_t)w * KN + k];
  s_out[k] = a;
  r_out[k] = K2[k] / a;
}
__device__ __forceinline__ void tail_reduce_adj(const double* __restrict__ part,
                                                const double* __restrict__ s,
                                                const double* __restrict__ r,
                                                double* __restrict__ dK2,
                                                double* __restrict__ dso) {
  const int k = threadIdx.x;
  double a = 0.0;
  for (int w = 0; w < NWG; ++w) a += part[(size_t)w * KN + k];
  dK2[k] += a / s[k];
  dso[k]  = -a * r[k] / s[k];
}

// ---- startup: state + counter zeroing --------------------------------------
__global__ __launch_bounds__(256)
void init_state_kernel(const float* __restrict__ w_in, float* __restrict__ wcur,
                       float* __restrict__ buf, unsigned* __restrict__ ctrs) {
  const int k = threadIdx.x;
  wcur[k] = w_in[k];
  buf[k]  = 0.0f;
  if (k < 64) ctrs[k] = 0u;
}

// ---- startup: E = exp(f/eps) AND s1[k] = sum_b E[b,k] in one pass ----------
__global__ __launch_bounds__(256)
void compute_E_s1_kernel(const float* __restrict__ f, double* __restrict__ E,
                         double* __restrict__ part, double* __restrict__ s1,
                         unsigned* __restrict__ ctr) {
  const int tid  = threadIdx.x;
  const int row0 = blockIdx.x * RPW;
  double acc = 0.0;
  for (int i = 0; i < RPW; ++i) {
    const size_t idx = (size_t)(row0 + i) * KN + tid;
    const double e = exp((double)f[idx] / 0.05);
    E[idx] = e;
    acc += e;
  }
  part[(size_t)blockIdx.x * KN + tid] = acc;
  if (block_is_last(ctr)) {
    const int k = threadIdx.x;
    double a = 0.0;
    for (int w = 0; w < NWG; ++w) a += part[(size_t)w * KN + k];
    s1[k] = a;
  }
}

// ---- per-iteration scalar chain: (grad clip + momentum + w step) then ------
// ---- K2 = softmax(w), r1 = K2/s1, dK2 = 0  (all in one 1-block kernel) -----
__global__ __launch_bounds__(256)
void update_w_kernel(float* __restrict__ wcur, float* __restrict__ mbuf,
                     float* __restrict__ sm32, double* __restrict__ dK2,
                     double* __restrict__ K2, const double* __restrict__ s1,
                     double* __restrict__ r1, int first) {
  __shared__ double red[KN];
  const int k = threadIdx.x;
  float wv = wcur[k];
  if (!first) {
    const float smk = sm32[k];
    const float dsm = (float)dK2[k];
    red[k] = (double)smk * (double)dsm; __syncthreads();
    for (int s2 = 128; s2; s2 >>= 1) { if (k < s2) red[k] += red[k + s2]; __syncthreads(); }
    const float dot = (float)red[0]; __syncthreads();
    float g = smk * (dsm - dot);                     // softmax VJP (d loss1 / dw)
    g += 5.0f * (smk - (1.0f / 256.0f)) / 256.0f;    // GAMMA * d(reg)/dw
    red[k] = (double)g * (double)g; __syncthreads();
    for (int s2 = 128; s2; s2 >>= 1) { if (k < s2) red[k] += red[k + s2]; __syncthreads(); }
    const float norm = (float)sqrt(red[0]); __syncthreads();
    const float scale = fminf(1.0f, 1.0f / (norm + 1e-6f));
    g *= scale;
    const float nb = 0.99f * mbuf[k] + g;            // momentum
    mbuf[k] = nb;
    wv -= 0.1f * nb;                                 // SGD step
    wcur[k] = wv;
  }
  // softmax(wv) in f32 (as the reference), then cast
  red[k] = (double)wv; __syncthreads();
  for (int s2 = 128; s2; s2 >>= 1) { if (k < s2) red[k] = fmax(red[k], red[k + s2]); __syncthreads(); }
  const float m = (float)red[0]; __syncthreads();
  const float e = expf(wv - m);
  red[k] = (double)e; __syncthreads();
  for (int s2 = 128; s2; s2 >>= 1) { if (k < s2) red[k] += red[k + s2]; __syncthreads(); }
  const float sv = e / (float)red[0];
  sm32[k] = sv;
  K2[k]   = (double)sv;
  dK2[k]  = 0.0;
  r1[k]   = (double)sv / s1[k];
}

// ---- fused forward pass: per row t=sum E*r -> c; column acc of E*c; -------
// ---- last block reduces partials into s_next, r_next = K2/s_next -----------
__global__ __launch_bounds__(256)
void fused_fwd_kernel(const double* __restrict__ E, const double* __restrict__ rin,
                      double* __restrict__ c_out, double* __restrict__ part,
                      const double* __restrict__ K2, double* __restrict__ s_out,
                      double* __restrict__ r_out, unsigned* __restrict__ ctr) {
  __shared__ double Eb[2][CHR * KN];
  __shared__ double rv[KN];
  __shared__ double cvv[CHR];
  const int tid = threadIdx.x, lane = tid & 31, wave = tid >> 5;
  const int row0 = blockIdx.x * RPW;
  rv[tid] = rin[tid];
  double acc = 0.0;
  #pragma unroll
  for (int rr = 0; rr < CHR; ++rr)
    async_b64(lds_addr_of(&Eb[0][rr * KN + tid]), &E[(size_t)(row0 + rr) * KN + tid]);
  for (int ch = 0; ch < NCH; ++ch) {
    __syncthreads();                               // prior phase2 done before overwrite
    const int cur = ch & 1;
    if (ch + 1 < NCH) {
      const int nb = cur ^ 1;
      #pragma unroll
      for (int rr = 0; rr < CHR; ++rr)
        async_b64(lds_addr_of(&Eb[nb][rr * KN + tid]),
                  &E[(size_t)(row0 + (ch + 1) * CHR + rr) * KN + tid]);
      wait_async_8();
    } else {
      wait_async_0();
    }
    __syncthreads();                               // all waves' chunk data resident
    {                                              // phase 1: wave w <-> chunk row w
      const double* rowp = &Eb[cur][wave * KN];
      double t = 0.0;
      #pragma unroll
      for (int j = 0; j < 8; ++j) { const int k = lane + 32 * j; t += rowp[k] * rv[k]; }
      for (int m = 16; m; m >>= 1) t += __shfl_xor(t, m, 32);
      if (lane == 0) {
        const double cb = 1.0 / (BND * t);
        cvv[wave] = cb;
        c_out[row0 + ch * CHR + wave] = cb;
      }
    }
    __syncthreads();                               // cvv published
    #pragma unroll
    for (int rr = 0; rr < CHR; ++rr)               // phase 2: column accumulation
      acc += Eb[cur][rr * KN + tid] * cvv[rr];
  }
  part[(size_t)blockIdx.x * KN + tid] = acc;
  if (block_is_last(ctr)) tail_reduce_fwd(part, K2, s_out, r_out);
}

// ---- fused backward head: c3, dt3 = Bn*c3^2*sum(E r3 f); acc E*(dt3-c3*f); -
// ---- last block: dK2 += a/s3, ds = -a*r3/s3 --------------------------------
__global__ __launch_bounds__(256)
void fused_bwd3_kernel(const double* __restrict__ E, const float* __restrict__ f,
                       const double* __restrict__ r3, double* __restrict__ part,
                       const double* __restrict__ s_lvl, const double* __restrict__ r_lvl,
                       double* __restrict__ dK2, double* __restrict__ dso,
                       unsigned* __restrict__ ctr) {
  __shared__ double Eb[2][CHR * KN];
  __shared__ float  Fb[2][CHR * KN];
  __shared__ double rv[KN];
  __shared__ double cvv[CHR];
  __shared__ double dvv[CHR];
  const int tid = threadIdx.x, lane = tid & 31, wave = tid >> 5;
  const int row0 = blockIdx.x * RPW;
  rv[tid] = r3[tid];
  double acc = 0.0;
  #pragma unroll
  for (int rr = 0; rr < CHR; ++rr) {
    async_b64(lds_addr_of(&Eb[0][rr * KN + tid]), &E[(size_t)(row0 + rr) * KN + tid]);
    async_b32(lds_addr_of(&Fb[0][rr * KN + tid]), &f[(size_t)(row0 + rr) * KN + tid]);
  }
  for (int ch = 0; ch < NCH; ++ch) {
    __syncthreads();
    const int cur = ch & 1;
    if (ch + 1 < NCH) {
      const int nb = cur ^ 1;
      #pragma unroll
      for (int rr = 0; rr < CHR; ++rr) {
        async_b64(lds_addr_of(&Eb[nb][rr * KN + tid]),
                  &E[(size_t)(row0 + (ch + 1) * CHR + rr) * KN + tid]);
        async_b32(lds_addr_of(&Fb[nb][rr * KN + tid]),
                  &f[(size_t)(row0 + (ch + 1) * CHR + rr) * KN + tid]);
      }
      wait_async_16();
    } else {
      wait_async_0();
    }
    __syncthreads();
    {
      const double* rowe = &Eb[cur][wave * KN];
      const float*  rowf = &Fb[cur][wave * KN];
      double t = 0.0, S = 0.0;
      #pragma unroll
      for (int j = 0; j < 8; ++j) {
        const int k = lane + 32 * j;
        const double er = rowe[k] * rv[k];
        t += er; S += er * (double)rowf[k];
      }
      for (int m = 16; m; m >>= 1) { t += __shfl_xor(t, m, 32); S += __shfl_xor(S, m, 32); }
      if (lane == 0) {
        const double cb = 1.0 / (BND * t);
        cvv[wave] = cb;
        dvv[wave] = BND * cb * cb * S;             // dt3 = -Bn*c3^2*dc3, dc3 = -S
      }
    }
    __syncthreads();
    #pragma unroll
    for (int rr = 0; rr < CHR; ++rr)
      acc += Eb[cur][rr * KN + tid] * (dvv[rr] - cvv[rr] * (double)Fb[cur][rr * KN + tid]);
  }
  part[(size_t)blockIdx.x * KN + tid] = acc;
  if (block_is_last(ctr)) tail_reduce_adj(part, s_lvl, r_lvl, dK2, dso);
}

// ---- fused backward tail: dt = -Bn*ci^2*sum(E*ds); acc E*dt; ---------------
// ---- last block: dK2 += a/s_lvl, ds_out = -a*r_lvl/s_lvl -------------------
__global__ __launch_bounds__(256)
void fused_bwd_kernel(const double* __restrict__ E, const double* __restrict__ dsv,
                      const double* __restrict__ ci, double* __restrict__ part,
                      const double* __restrict__ s_lvl, const double* __restrict__ r_lvl,
                      double* __restrict__ dK2, double* __restrict__ dso,
                      unsigned* __restrict__ ctr) {
  __shared__ double Eb[2][CHR * KN];
  __shared__ double rv[KN];
  __shared__ double dvv[CHR];
  const int tid = threadIdx.x, lane = tid & 31, wave = tid >> 5;
  const int row0 = blockIdx.x * RPW;
  rv[tid] = dsv[tid];                              // safe: consumed before any block rewrites dso
  double acc = 0.0;
  #pragma unroll
  for (int rr = 0; rr < CHR; ++rr)
    async_b64(lds_addr_of(&Eb[0][rr * KN + tid]), &E[(size_t)(row0 + rr) * KN + tid]);
  for (int ch = 0; ch < NCH; ++ch) {
    __syncthreads();
    const int cur = ch & 1;
    if (ch + 1 < NCH) {
      const int nb = cur ^ 1;
      #pragma unroll
      for (int rr = 0; rr < CHR; ++rr)
        async_b64(lds_addr_of(&Eb[nb][rr * KN + tid]),
                  &E[(size_t)(row0 + (ch + 1) * CHR + rr) * KN + tid]);
      wait_async_8();
    } else {
      wait_async_0();
    }
    __syncthreads();
    {
      const double* rowp = &Eb[cur][wave * KN];
      double S = 0.0;
      #pragma unroll
      for (int j = 0; j < 8; ++j) { const int k = lane + 32 * j; S += rowp[k] * rv[k]; }
      for (int m = 16; m; m >>= 1) S += __shfl_xor(S, m, 32);
      if (lane == 0) {
        const double cb = ci[row0 + ch * CHR + wave];
        dvv[wave] = -BND * cb * cb * S;
      }
    }
    __syncthreads();
    #pragma unroll
    for (int rr = 0; rr < CHR; ++rr)
      acc += Eb[cur][rr * KN + tid] * dvv[rr];
  }
  part[(size_t)blockIdx.x * KN + tid] = acc;
  if (block_is_last(ctr)) tail_reduce_adj(part, s_lvl, r_lvl, dK2, dso);
}

// ---- fused final pass: c3 per row, then Q = Bn*E*r3*c3 -> f32 out ----------
__global__ __launch_bounds__(256)
void fused_q_kernel(const double* __restrict__ E, const double* __restrict__ r3,
                    float* __restrict__ out) {
  __shared__ double Eb[2][CHR * KN];
  __shared__ double rv[KN];
  __shared__ double cvv[CHR];
  const int tid = threadIdx.x, lane = tid & 31, wave = tid >> 5;
  const int row0 = blockIdx.x * RPW;
  rv[tid] = r3[tid];
  #pragma unroll
  for (int rr = 0; rr < CHR; ++rr)
    async_b64(lds_addr_of(&Eb[0][rr * KN + tid]), &E[(size_t)(row0 + rr) * KN + tid]);
  for (int ch = 0; ch < NCH; ++ch) {
    __syncthreads();
    const int cur = ch & 1;
    if (ch + 1 < NCH) {
      const int nb = cur ^ 1;
      #pragma unroll
      for (int rr = 0; rr < CHR; ++rr)
        async_b64(lds_addr_of(&Eb[nb][rr * KN + tid]),
                  &E[(size_t)(row0 + (ch + 1) * CHR + rr) * KN + tid]);
      wait_async_8();
    } else {
      wait_async_0();
    }
    __syncthreads();
    {
      const double* rowp = &Eb[cur][wave * KN];
      double t = 0.0;
      #pragma unroll
      for (int j = 0; j < 8; ++j) { const int k = lane + 32 * j; t += rowp[k] * rv[k]; }
      for (int m = 16; m; m >>= 1) t += __shfl_xor(t, m, 32);
      if (lane == 0) cvv[wave] = 1.0 / (BND * t);
    }
    __syncthreads();
    #pragma unroll
    for (int rr = 0; rr < CHR; ++rr) {
      const size_t b = (size_t)(row0 + ch * CHR + rr);
      out[b * KN + tid] = (float)(BND * Eb[cur][rr * KN + tid] * rv[tid] * cvv[rr]);
    }
  }
}

} // namespace

extern "C" void kernel_launch(void* const* d_in, const int* in_sizes, int n_in,
                              void* d_out, int out_size, void* d_ws, size_t ws_size,
                              hipStream_t stream) {
  (void)in_sizes; (void)n_in; (void)out_size; (void)ws_size;
  const float* f  = (const float*)d_in[0];   // [B,K] f32
  const float* w0 = (const float*)d_in[1];   // [1,K] f32

  char* ws = (char*)d_ws;
  size_t off = 0;
  auto take = [&](size_t bytes) { size_t o = off; off += (bytes + 255) & ~(size_t)255; return o; };
  double*   E    = (double*)(ws + take((size_t)BN * KN * sizeof(double)));   // 33.5MB, L2-resident
  double*   part = (double*)(ws + take((size_t)NWG * KN * sizeof(double)));
  double*   s    = (double*)(ws + take(3 * KN * sizeof(double)));            // s1, s2, s3
  double*   r    = (double*)(ws + take(3 * KN * sizeof(double)));            // r1, r2, r3
  double*   c    = (double*)(ws + take((size_t)2 * BN * sizeof(double)));    // c1, c2
  double*   K2   = (double*)(ws + take(KN * sizeof(double)));
  double*   dK2  = (double*)(ws + take(KN * sizeof(double)));
  double*   ds   = (double*)(ws + take(KN * sizeof(double)));
  float*    sm32 = (float*)(ws + take(KN * sizeof(float)));
  float*    wcur = (float*)(ws + take(KN * sizeof(float)));
  float*    mbuf = (float*)(ws + take(KN * sizeof(float)));
  unsigned* ctrs = (unsigned*)(ws + take(64 * sizeof(unsigned)));

  int ci = 0;  // one fresh counter per big-pass instance (zeroed in init_state)

  // startup: state + counters; E = exp(f/eps) fused with s1 = sum_b E[b,k]
  init_state_kernel<<<1, KN, 0, stream>>>(w0, wcur, mbuf, ctrs);
  compute_E_s1_kernel<<<NWG, KN, 0, stream>>>(f, E, part, s + 0 * KN, ctrs + ci++);

  for (int it = 0; it < 10; ++it) {
    // (clip+momentum+step for prev iter) then K2=softmax(w), r1=K2/s1, dK2=0
    update_w_kernel<<<1, KN, 0, stream>>>(wcur, mbuf, sm32, dK2, K2, s + 0 * KN,
                                          r + 0 * KN, it == 0 ? 1 : 0);

    // forward (fused: c_i + s_{i+1}; tail reduce -> r_{i+1})
    fused_fwd_kernel<<<NWG, KN, 0, stream>>>(E, r + 0 * KN, c + (size_t)0 * BN, part,
                                             K2, s + 1 * KN, r + 1 * KN, ctrs + ci++);
    fused_fwd_kernel<<<NWG, KN, 0, stream>>>(E, r + 1 * KN, c + (size_t)1 * BN, part,
                                             K2, s + 2 * KN, r + 2 * KN, ctrs + ci++);

    if (it == 9) {  // last step: fused c3 + Q write (Q uses pre-update w)
      fused_q_kernel<<<NWG, KN, 0, stream>>>(E, r + 2 * KN, (float*)d_out);
      break;
    }

    // backward (fused passes; tail reduces produce ds / accumulate dK2)
    fused_bwd3_kernel<<<NWG, KN, 0, stream>>>(E, f, r + 2 * KN, part,
                                              s + 2 * KN, r + 2 * KN, dK2, ds, ctrs + ci++);
    fused_bwd_kernel<<<NWG, KN, 0, stream>>>(E, ds, c + (size_t)1 * BN, part,
                                             s + 1 * KN, r + 1 * KN, dK2, ds, ctrs + ci++);
    fused_bwd_kernel<<<NWG, KN, 0, stream>>>(E, ds, c + (size_t)0 * BN, part,
                                             s + 0 * KN, r + 0 * KN, dK2, ds, ctrs + ci++);
  }
}